// AttentiveStatisticsPooling_9543417331966
// MI455X (gfx1250) — compile-verified
//
#include <hip/hip_runtime.h>
#include <math.h>

// Problem constants (match reference)
#define BB 64
#define TT 1024
#define DD 512
#define EPS_F 1e-5f

typedef __attribute__((ext_vector_type(2))) float v2f;
typedef __attribute__((ext_vector_type(8))) float v8f;

#define TM 32          // rows per workgroup in pass 1
#define KC 64          // K-chunk staged in LDS
#define LDS_STRIDE 68  // 64 + 4 pad: keeps 16B alignment of rows, avoids bank conflicts

// ---------------------------------------------------------------------------
// Pass 1: logits[b*T + t] = sum_e att[e] * tanh( sum_d xs[row][d]*W[e][d] + b[e] )
// Fused GEMM (f32 WMMA) + tanh + dot(att). No h materialization.
// ---------------------------------------------------------------------------
__global__ __launch_bounds__(256) void asp_pass1_logits(
    const float* __restrict__ xs, const float* __restrict__ W,
    const float* __restrict__ bias, const float* __restrict__ att,
    float* __restrict__ logits)
{
    __shared__ float tile[TM * LDS_STRIDE];  // 32 x 68 floats = 8704 B
    __shared__ float red[TM];

    const int tid  = threadIdx.x;
    const int wave = tid >> 5;   // 0..7
    const int lane = tid & 31;
    const int lhalf = lane >> 4; // 0 or 1 (K-pair select for WMMA A/B frags)
    const int l16   = lane & 15;

    const int row0 = blockIdx.x * TM;  // flat row base into (B*T)
    const int e0   = wave * 64;        // this wave's output-column base

    if (tid < TM) red[tid] = 0.0f;

    // accumulators: 2 row-subtiles (16 rows each) x 4 N-tiles (16 cols each)
    v8f acc[2][4];
    #pragma unroll
    for (int s = 0; s < 2; ++s)
        #pragma unroll
        for (int n = 0; n < 4; ++n)
            #pragma unroll
            for (int v = 0; v < 8; ++v)
                acc[s][n][v] = 0.0f;

    for (int kc = 0; kc < DD; kc += KC) {
        __syncthreads();  // protect tile (readers of previous chunk done; also fences red init)
        // cooperative stage: xs[row0..row0+31][kc..kc+63] -> LDS
        // 32 rows x 16 float4 = 512 slots, 256 threads x 2, fully coalesced
        #pragma unroll
        for (int i = 0; i < 2; ++i) {
            const int slot = tid + i * 256;
            const int r  = slot >> 4;
            const int c4 = slot & 15;
            const float4 v = *reinterpret_cast<const float4*>(
                xs + (size_t)(row0 + r) * DD + kc + c4 * 4);
            *reinterpret_cast<float4*>(&tile[r * LDS_STRIDE + c4 * 4]) = v;
        }
        __syncthreads();

        // speculative prefetch of next K-chunk of W for this wave's column slice
        if (kc + KC < DD)
            __builtin_prefetch(W + (size_t)e0 * DD + (kc + KC), 0, 1);

        // 16 k-steps of K=4
        #pragma unroll 4
        for (int ks = 0; ks < KC / 4; ++ks) {
            const int kb = ks * 4 + 2 * lhalf;  // lane's K base (A/B VGPR layout)

            // A fragments (16x4 f32): lane l16 = row, lhalf selects K-pair
            v2f a[2];
            #pragma unroll
            for (int s = 0; s < 2; ++s)
                a[s] = *reinterpret_cast<const v2f*>(
                    &tile[(s * 16 + l16) * LDS_STRIDE + kb]);

            // B fragments (4x16 f32): B[k][j] = W[e0+j][d], lane l16 = column j
            #pragma unroll
            for (int n = 0; n < 4; ++n) {
                const v2f bfrag = *reinterpret_cast<const v2f*>(
                    W + (size_t)(e0 + n * 16 + l16) * DD + kc + kb);
                #pragma unroll
                for (int s = 0; s < 2; ++s)
                    acc[s][n] = __builtin_amdgcn_wmma_f32_16x16x4_f32(
                        false, a[s], false, bfrag, (short)0, acc[s][n],
                        false, false);
            }
        }
    }

    // Epilogue: tanh + att weighting, reduce over columns.
    // C/D layout: vgpr v, lanes 0-15 -> row (subtile)*16 + v, lanes 16-31 -> +8.
    #pragma unroll
    for (int s = 0; s < 2; ++s) {
        float part[8];
        #pragma unroll
        for (int v = 0; v < 8; ++v) part[v] = 0.0f;
        #pragma unroll
        for (int n = 0; n < 4; ++n) {
            const int e = e0 + n * 16 + l16;
            const float be = bias[e];
            const float ae = att[e];
            #pragma unroll
            for (int v = 0; v < 8; ++v)
                part[v] += tanhf(acc[s][n][v] + be) * ae;
        }
        #pragma unroll
        for (int v = 0; v < 8; ++v) {
            float p = part[v];
            p += __shfl_xor(p, 1, 32);
            p += __shfl_xor(p, 2, 32);
            p += __shfl_xor(p, 4, 32);
            p += __shfl_xor(p, 8, 32);   // stays within each 16-lane half
            if (l16 == 0)
                atomicAdd(&red[s * 16 + v + 8 * lhalf], p);  // ds_add_f32
        }
    }
    __syncthreads();
    if (tid < TM) logits[row0 + tid] = red[tid];
}

// ---------------------------------------------------------------------------
// Pass 2: per batch: masked softmax over T, then mu/std over xs columns.
// ---------------------------------------------------------------------------
__global__ __launch_bounds__(256) void asp_pass2_pool(
    const float* __restrict__ xs, const float* __restrict__ mask,
    const float* __restrict__ logits, float* __restrict__ out)
{
    __shared__ float wbuf[TT];
    __shared__ float sred[8];

    const int b    = blockIdx.x;
    const int tid  = threadIdx.x;
    const int lane = tid & 31;
    const int wave = tid >> 5;
    const float NEG_INF = -__builtin_inff();

    // masked logits, thread handles t = tid + i*256
    float lv[4];
    float lmax = NEG_INF;
    #pragma unroll
    for (int i = 0; i < 4; ++i) {
        const int t = tid + i * 256;
        const float m = mask[b * TT + t];
        const float v = (m > 0.0f) ? logits[b * TT + t] : NEG_INF;
        lv[i] = v;
        lmax = fmaxf(lmax, v);
    }
    #pragma unroll
    for (int off = 16; off > 0; off >>= 1)
        lmax = fmaxf(lmax, __shfl_xor(lmax, off, 32));
    if (lane == 0) sred[wave] = lmax;
    __syncthreads();
    if (tid == 0) {
        float m = sred[0];
        for (int i = 1; i < 8; ++i) m = fmaxf(m, sred[i]);
        sred[0] = m;
    }
    __syncthreads();
    const float gmax = sred[0];
    __syncthreads();  // sred reused below

    float ev[4];
    float lsum = 0.0f;
    #pragma unroll
    for (int i = 0; i < 4; ++i) {
        const float e = (lv[i] == NEG_INF) ? 0.0f : __expf(lv[i] - gmax);
        ev[i] = e;
        lsum += e;
    }
    #pragma unroll
    for (int off = 16; off > 0; off >>= 1)
        lsum += __shfl_xor(lsum, off, 32);
    if (lane == 0) sred[wave] = lsum;
    __syncthreads();
    if (tid == 0) {
        float ssum = 0.0f;
        for (int i = 0; i < 8; ++i) ssum += sred[i];
        sred[0] = ssum;
    }
    __syncthreads();
    const float inv = 1.0f / sred[0];
    #pragma unroll
    for (int i = 0; i < 4; ++i) wbuf[tid + i * 256] = ev[i] * inv;
    __syncthreads();

    // weighted stats: thread owns columns d = 2*tid, 2*tid+1 (coalesced float2)
    float2 mu = make_float2(0.0f, 0.0f);
    float2 m2 = make_float2(0.0f, 0.0f);
    const float* xb = xs + (size_t)b * TT * DD + 2 * tid;
    for (int t = 0; t < TT; ++t) {
        const float wv = wbuf[t];  // LDS broadcast
        const float2 x = *reinterpret_cast<const float2*>(xb + (size_t)t * DD);
        mu.x += wv * x.x;           mu.y += wv * x.y;
        m2.x += wv * x.x * x.x;     m2.y += wv * x.y * x.y;
    }
    float2 rh;
    rh.x = sqrtf(fmaxf(m2.x - mu.x * mu.x, EPS_F));
    rh.y = sqrtf(fmaxf(m2.y - mu.y * mu.y, EPS_F));

    float* ob = out + (size_t)b * (2 * DD);
    *reinterpret_cast<float2*>(ob + 2 * tid) = mu;        // out[b][0:512]   = mu
    *reinterpret_cast<float2*>(ob + DD + 2 * tid) = rh;   // out[b][512:1024]= std
}

// ---------------------------------------------------------------------------
extern "C" void kernel_launch(void* const* d_in, const int* in_sizes, int n_in,
                              void* d_out, int out_size, void* d_ws, size_t ws_size,
                              hipStream_t stream)
{
    (void)in_sizes; (void)n_in; (void)out_size; (void)ws_size;
    const float* xs   = (const float*)d_in[0];  // (B, T, D)
    const float* mask = (const float*)d_in[1];  // (B, T)
    const float* W    = (const float*)d_in[2];  // (D, D)
    const float* bias = (const float*)d_in[3];  // (D,)
    const float* att  = (const float*)d_in[4];  // (D,)
    float* out    = (float*)d_out;              // (B, 2*D)
    float* logits = (float*)d_ws;               // (B*T) f32 scratch = 256 KB

    const int nrowtiles = (BB * TT) / TM;  // 2048 workgroups
    asp_pass1_logits<<<nrowtiles, 256, 0, stream>>>(xs, W, bias, att, logits);
    asp_pass2_pool<<<BB, 256, 0, stream>>>(xs, mask, logits, out);
}